// ChebySemi_4509715660933
// MI455X (gfx1250) — compile-verified
//
#include <hip/hip_runtime.h>

typedef float v2f __attribute__((ext_vector_type(2)));
typedef float v8f __attribute__((ext_vector_type(8)));

#define MDIM 512
#define PS   514                 // padded row length (512 + 2 halo)
#define PROWS 516                // rows allocated: 514 + 2 slack (zeroed, K-pad reads)
#define SAMPLE_P (PS * PROWS)    // floats per padded sample buffer

__constant__ float c_roots[15] = {
     0.9945218954f,  0.9510565163f,  0.8660254038f,  0.7431448255f,
     0.5877852523f,  0.4067366431f,  0.2079116908f,  0.0f,
    -0.2079116908f, -0.4067366431f, -0.5877852523f, -0.7431448255f,
    -0.8660254038f, -0.9510565163f, -0.9945218954f };

__device__ __forceinline__ float pick3(int d, float k0, float k1, float k2) {
    float v = 0.0f;
    v = (d == 0) ? k0 : v;
    v = (d == 1) ? k1 : v;
    v = (d == 2) ? k2 : v;
    return v;
}

// Banded A in WMMA A-layout (16x4 per chunk): vgpr0 K={k0,k0+2}, vgpr1 K={k0+1,k0+3};
// A[m][q] = ker[q-m][j] * inv_s for q-m in {0,1,2}, else 0 (K padded 18->20).
__device__ __forceinline__ void build_a(const float (&kc)[9], float inv_s, bool hi, int mrow,
                                        float (&aLo)[3][5], float (&aHi)[3][5]) {
#pragma unroll
    for (int j = 0; j < 3; ++j) {
        const float kj0 = kc[0 + j] * inv_s;
        const float kj1 = kc[3 + j] * inv_s;
        const float kj2 = kc[6 + j] * inv_s;
#pragma unroll
        for (int ch = 0; ch < 5; ++ch) {
            const int q0 = 4 * ch + (hi ? 2 : 0);
            const int q1 = 4 * ch + (hi ? 3 : 1);
            aLo[j][ch] = pick3(q0 - mrow, kj0, kj1, kj2);
            aHi[j][ch] = pick3(q1 - mrow, kj0, kj1, kj2);
        }
    }
}

// Padded-image borders: top row / left col = 0, bottom row / right col = bval
// ("1.0" compensated to m when A carries inv_s). Corners match the reference
// (the value-1 edges win everywhere they overlap the zero edges).
__device__ __forceinline__ void fill_borders(float* __restrict__ p, float bval) {
    for (int i = threadIdx.x; i < PS; i += blockDim.x) {
        p[i]                 = (i == PS - 1) ? bval : 0.0f;  // top row
        p[(PS - 1) * PS + i] = bval;                          // bottom row
        p[i * PS]            = (i == PS - 1) ? bval : 0.0f;  // left col
        p[i * PS + PS - 1]   = bval;                          // right col
    }
}

__device__ __forceinline__ void copy_interior(const float* __restrict__ s, float* __restrict__ p) {
    for (int i = threadIdx.x; i < MDIM * MDIM; i += blockDim.x) {
        const int r = i >> 9, c = i & (MDIM - 1);
        p[(r + 1) * PS + (c + 1)] = s[i];
    }
}

// One full 512x512 sweep. Wave w owns a 16-row stripe; 32 tiles of 16x16 each;
// tile = 15 x V_WMMA_F32_16X16X4_F32, all B loads are immediate offsets off one
// base address into the padded src (no clamps, no selects, EXEC stays all-ones).
template <bool IS_POWER, int DSTRIDE>
__device__ __forceinline__ float sweep(const float* __restrict__ srcP,   // padded base
                                       float* __restrict__ dst,          // interior origin / unpadded out
                                       const float* __restrict__ fsrc,
                                       float tau,
                                       const float (&aLo)[3][5], const float (&aHi)[3][5]) {
    const int lane = threadIdx.x & 31;
    const int wave = threadIdx.x >> 5;
    const bool hi  = lane >= 16;
    const int col  = lane & 15;
    const int R    = wave * 16;
    float lmax = 0.0f;

#pragma unroll 1
    for (int ct = 0; ct < 32; ++ct) {
        const int C = ct * 16;
        __builtin_prefetch(&srcP[R * PS + C + 16 + col], 0, 3);  // next tile -> WGP$
        const int base0 = (R + (hi ? 2 : 0)) * PS + C + col;     // K rows {k0, k0+2}
        const int base1 = (R + (hi ? 3 : 1)) * PS + C + col;     // K rows {k0+1, k0+3}
        v8f acc = {0.f, 0.f, 0.f, 0.f, 0.f, 0.f, 0.f, 0.f};
#pragma unroll
        for (int j = 0; j < 3; ++j) {
#pragma unroll
            for (int ch = 0; ch < 5; ++ch) {
                v2f bm, am;
                bm.x = srcP[base0 + ch * (4 * PS) + j];
                bm.y = srcP[base1 + ch * (4 * PS) + j];
                am.x = aLo[j][ch];
                am.y = aHi[j][ch];
                acc = __builtin_amdgcn_wmma_f32_16x16x4_f32(
                    false, am, false, bm, (short)0, acc, false, false);
            }
        }
        // D layout: vgpr v -> (row R+v, lanes 0-15) and (row R+v+8, lanes 16-31)
#pragma unroll
        for (int v = 0; v < 8; ++v) {
            const int row = R + v + (hi ? 8 : 0);
            const float y = acc[v];
            if (IS_POWER) {
                dst[row * DSTRIDE + C + col] = y;
                lmax = fmaxf(lmax, fabsf(y));
            } else {
                const float xv = srcP[(row + 1) * PS + (C + col + 1)];
                dst[row * DSTRIDE + C + col] = xv + tau * (fsrc[row * MDIM + C + col] - y);
            }
        }
    }
    return lmax;
}

__global__ __launch_bounds__(1024, 1)
void cheb_semi_kernel(const float* __restrict__ x_in, const float* __restrict__ f_in,
                      const float* __restrict__ kA,  const float* __restrict__ u0,
                      float* __restrict__ x_out, float* __restrict__ ws) {
    const int b = blockIdx.x;
    const size_t off = (size_t)b * MDIM * MDIM;
    const float* xb  = x_in + off;
    const float* fb  = f_in + off;
    const float* u0b = u0   + off;
    float* outb = x_out + off;
    float* pA = ws + (size_t)(2 * b) * SAMPLE_P;
    float* pB = pA + SAMPLE_P;

    float kc[9];
#pragma unroll
    for (int i = 0; i < 9; ++i) kc[i] = kA[b * 9 + i];

    const int lane = threadIdx.x & 31;
    const bool hi  = lane >= 16;
    const int mrow = lane & 15;
    float aLo[3][5], aHi[3][5];

    __shared__ unsigned s_max;

    // ---- init: u0 -> padded pA (border 1 since inv_s=1); zero slack rows ----
    copy_interior(u0b, pA);
    fill_borders(pA, 1.0f);
    for (int i = threadIdx.x; i < 2 * PS; i += blockDim.x) {
        pA[PS * PS + i] = 0.0f;
        pB[PS * PS + i] = 0.0f;
    }
    __threadfence();
    __syncthreads();

    // ---- power iteration: 20 sweeps, padded ping-pong, track max|y| ----
    // src holds *unnormalized* previous y; 1/m folded into A; pad-1 border
    // compensated by writing border cells = m after each sweep.
    const float* src = pA;
    float mval = 1.0f;
    for (int it = 0; it < 20; ++it) {
        float* dst = (it & 1) ? pA : pB;
        build_a(kc, 1.0f / mval, hi, mrow, aLo, aHi);
        if (threadIdx.x == 0) s_max = 0u;
        __syncthreads();
        const float lmax = sweep<true, PS>(src, dst + PS + 1, fb, 0.0f, aLo, aHi);
        atomicMax(&s_max, __float_as_uint(lmax));   // |y| >= 0: uint order == float order
        __threadfence();
        __syncthreads();
        mval = __uint_as_float(s_max);
        fill_borders(dst, mval);                    // border for next sweep's read
        __threadfence();
        __syncthreads();                            // also protects s_max reset
        src = dst;
    }

    // ---- Chebyshev: x -> padded pA; 15 sweeps; A unscaled so borders are 1 ----
    copy_interior(xb, pA);
    fill_borders(pA, 1.0f);
    fill_borders(pB, 1.0f);
    __threadfence();
    __syncthreads();
    build_a(kc, 1.0f, hi, mrow, aLo, aHi);

    const float* xs = pA;
    for (int k = 0; k < 14; ++k) {
        float* xd = (k & 1) ? pA : pB;              // k=13 leaves result in pA
        const float tau = 2.0f / (1.5f * mval + 0.5f * mval * c_roots[k]);
        sweep<false, PS>(xs, xd + PS + 1, fb, tau, aLo, aHi);
        __threadfence();
        __syncthreads();
        xs = xd;
    }
    // final (15th) sweep writes the unpadded output directly
    {
        const float tau = 2.0f / (1.5f * mval + 0.5f * mval * c_roots[14]);
        sweep<false, MDIM>(xs, outb, fb, tau, aLo, aHi);
    }
}

extern "C" void kernel_launch(void* const* d_in, const int* in_sizes, int n_in,
                              void* d_out, int out_size, void* d_ws, size_t ws_size,
                              hipStream_t stream) {
    const float* x  = (const float*)d_in[0];
    const float* f  = (const float*)d_in[1];
    const float* kA = (const float*)d_in[2];
    const float* u0 = (const float*)d_in[3];
    float* out = (float*)d_out;
    float* ws  = (float*)d_ws;   // needs >= 64 * 2 * 516*514 * 4 B ~= 130 MiB

    const int B = in_sizes[2] / 9;   // 64 samples
    cheb_semi_kernel<<<dim3(B), dim3(1024), 0, stream>>>(x, f, kA, u0, out, ws);
}